// GCNGAT_85727547228226
// MI455X (gfx1250) — compile-verified
//
#include <hip/hip_runtime.h>
#include <math.h>

#define N_NODES 50000
#define N_EDGES 800000
#define TOT_E   (N_EDGES + N_NODES)
#define IN_DIM  128
#define HID     128
#define HEADS   2
#define GO      64
#define OUT_DIM 64

typedef float v2f __attribute__((ext_vector_type(2)));
typedef float v8f __attribute__((ext_vector_type(8)));

static inline int cdiv(long long a, long long b) { return (int)((a + b - 1) / b); }

// ---------------------------------------------------------------------------
// Order-preserving float <-> uint mapping for atomic segment-max.
// Sentinel 0u is strictly below every encoded float (incl. -inf).
// ---------------------------------------------------------------------------
__device__ __forceinline__ unsigned fmax_encode(float f) {
  unsigned u = __float_as_uint(f);
  return (u & 0x80000000u) ? ~u : (u | 0x80000000u);
}
__device__ __forceinline__ float fmax_decode(unsigned u) {
  unsigned b = (u & 0x80000000u) ? (u ^ 0x80000000u) : ~u;
  return __uint_as_float(b);
}

__device__ __forceinline__ float lrelu02(float v) { return v > 0.0f ? v : 0.2f * v; }

// ---------------------------------------------------------------------------
// FP32 WMMA GEMM: C[M, ncols] = A[M, K] * B[K, ncols], row-major.
// One wave per 16x16 C tile; blockIdx.x = row tile; wave id = column tile.
// Uses V_WMMA_F32_16X16X4_F32 (K-step 4, full fp32 precision).
// A frag:  lane&15 -> M row, (lane>>4)*2 + {0,1} -> K       (ISA 16x4 layout)
// B frag:  lane&15 -> N col, (lane>>4)*2 + {0,1} -> K       (4x16, rows across lanes)
// C frag:  vgpr r, half h -> M = r + 8h, N = lane&15
// ---------------------------------------------------------------------------
__global__ void gemm_f32_wmma(const float* __restrict__ A,
                              const float* __restrict__ B,
                              float* __restrict__ C,
                              int K, int ncols) {
  const int mt   = blockIdx.x;
  const int wave = threadIdx.x >> 5;
  const int lane = threadIdx.x & 31;
  if (wave >= (ncols >> 4)) return;          // uniform per wave: WMMA EXEC stays all-1s
  const int r16   = lane & 15;
  const int half  = lane >> 4;               // 0 | 1
  const int khalf = half * 2;                // 0 | 2
  const int row   = mt * 16 + r16;
  const int col   = wave * 16 + r16;
  const float* __restrict__ arow = A + (size_t)row * K;

  v8f acc = {};
  for (int k = 0; k < K; k += 4) {
    v2f a, b;
    a.x = arow[k + khalf];
    a.y = arow[k + khalf + 1];
    b.x = B[(size_t)(k + khalf)     * ncols + col];
    b.y = B[(size_t)(k + khalf + 1) * ncols + col];
    acc = __builtin_amdgcn_wmma_f32_16x16x4_f32(false, a, false, b,
                                                (short)0, acc, false, false);
  }
  const int rbase = mt * 16 + half * 8;
  for (int r = 0; r < 8; ++r)
    C[(size_t)(rbase + r) * ncols + col] = acc[r];
}

// ---------------------------------------------------------------------------
// Degree (over dst, incl. self-loops) then in-place inv-sqrt.
// ---------------------------------------------------------------------------
__global__ void deg_count(const int* __restrict__ dst, float* __restrict__ deg) {
  int e = blockIdx.x * blockDim.x + threadIdx.x;
  if (e >= TOT_E) return;
  int d = (e < N_EDGES) ? dst[e] : (e - N_EDGES);
  atomicAdd(&deg[d], 1.0f);
}

__global__ void inv_sqrt_k(float* __restrict__ deg) {
  int n = blockIdx.x * blockDim.x + threadIdx.x;
  if (n >= N_NODES) return;
  float v = deg[n];
  deg[n] = v > 0.0f ? rsqrtf(v) : 0.0f;
}

// ---------------------------------------------------------------------------
// GCN edge scatter: out[dst] += h[src] * inv[src]*inv[dst]. F/4 lanes per edge,
// float4 gather (coalesced 4*F bytes per edge row), fp32 global atomics.
// ---------------------------------------------------------------------------
template <int F>
__global__ void gcn_scatter(const float* __restrict__ h, const float* __restrict__ inv,
                            const int* __restrict__ src, const int* __restrict__ dst,
                            float* __restrict__ out) {
  const int tpe = F / 4;
  unsigned gtid = blockIdx.x * blockDim.x + threadIdx.x;
  int e    = (int)(gtid / tpe);
  int lane = (int)(gtid % tpe);
  if (e >= TOT_E) return;
  int s, d;
  if (e < N_EDGES) { s = src[e]; d = dst[e]; } else { s = d = e - N_EDGES; }
  float coef = inv[s] * inv[d];
  float4 v = reinterpret_cast<const float4*>(h + (size_t)s * F)[lane];
  float* o = out + (size_t)d * F + lane * 4;
  atomicAdd(o + 0, v.x * coef);
  atomicAdd(o + 1, v.y * coef);
  atomicAdd(o + 2, v.z * coef);
  atomicAdd(o + 3, v.w * coef);
}

// ---------------------------------------------------------------------------
// Elementwise bias + activation (mode 0: +b; 1: relu(+b)).
// ---------------------------------------------------------------------------
template <int F, int MODE>
__global__ void bias_act(float* __restrict__ buf, const float* __restrict__ b) {
  unsigned t = blockIdx.x * blockDim.x + threadIdx.x;
  if (t >= (unsigned)N_NODES * F) return;
  float v = buf[t] + b[t % F];
  if (MODE == 1) v = v > 0.0f ? v : 0.0f;
  buf[t] = v;
}

// ---------------------------------------------------------------------------
// GAT pieces
// ---------------------------------------------------------------------------
__global__ void attn_coeff(const float* __restrict__ hg,
                           const float* __restrict__ att_src,
                           const float* __restrict__ att_dst,
                           float* __restrict__ as_, float* __restrict__ ad_) {
  int t = blockIdx.x * blockDim.x + threadIdx.x;
  if (t >= N_NODES * HEADS) return;
  int n = t / HEADS, h = t % HEADS;
  const float* hv = hg + (size_t)n * (HEADS * GO) + h * GO;
  const float* w1 = att_src + h * GO;
  const float* w2 = att_dst + h * GO;
  float s1 = 0.0f, s2 = 0.0f;
  for (int f = 0; f < GO; ++f) { float x = hv[f]; s1 += x * w1[f]; s2 += x * w2[f]; }
  as_[t] = s1;
  ad_[t] = s2;
}

__global__ void gat_edge_max(const float* __restrict__ as_, const float* __restrict__ ad_,
                             const int* __restrict__ src, const int* __restrict__ dst,
                             unsigned* __restrict__ m) {
  int e = blockIdx.x * blockDim.x + threadIdx.x;
  if (e >= TOT_E) return;
  int s, d;
  if (e < N_EDGES) { s = src[e]; d = dst[e]; } else { s = d = e - N_EDGES; }
  for (int h = 0; h < HEADS; ++h) {
    float v = lrelu02(as_[s * HEADS + h] + ad_[d * HEADS + h]);
    atomicMax(&m[d * HEADS + h], fmax_encode(v));
  }
}

__global__ void gat_edge_sum(const float* __restrict__ as_, const float* __restrict__ ad_,
                             const int* __restrict__ src, const int* __restrict__ dst,
                             const unsigned* __restrict__ m, float* __restrict__ denom) {
  int e = blockIdx.x * blockDim.x + threadIdx.x;
  if (e >= TOT_E) return;
  int s, d;
  if (e < N_EDGES) { s = src[e]; d = dst[e]; } else { s = d = e - N_EDGES; }
  for (int h = 0; h < HEADS; ++h) {
    float v  = lrelu02(as_[s * HEADS + h] + ad_[d * HEADS + h]);
    float ex = expf(v - fmax_decode(m[d * HEADS + h]));
    atomicAdd(&denom[d * HEADS + h], ex);
  }
}

__global__ void gat_scatter(const float* __restrict__ hg,
                            const float* __restrict__ as_, const float* __restrict__ ad_,
                            const unsigned* __restrict__ m, const float* __restrict__ denom,
                            const int* __restrict__ src, const int* __restrict__ dst,
                            float* __restrict__ out) {
  unsigned gtid = blockIdx.x * blockDim.x + threadIdx.x;
  int e = (int)(gtid >> 5);
  int lane = (int)(gtid & 31);            // 32 lanes x 4 floats = HEADS*GO = 128
  if (e >= TOT_E) return;
  int s, d;
  if (e < N_EDGES) { s = src[e]; d = dst[e]; } else { s = d = e - N_EDGES; }
  int h = lane >> 4;                      // lane*4 / 64
  float v     = lrelu02(as_[s * HEADS + h] + ad_[d * HEADS + h]);
  float alpha = expf(v - fmax_decode(m[d * HEADS + h])) / denom[d * HEADS + h];
  float4 x = reinterpret_cast<const float4*>(hg + (size_t)s * (HEADS * GO))[lane];
  float* o = out + (size_t)d * (HEADS * GO) + lane * 4;
  atomicAdd(o + 0, x.x * alpha);
  atomicAdd(o + 1, x.y * alpha);
  atomicAdd(o + 2, x.z * alpha);
  atomicAdd(o + 3, x.w * alpha);
}

__global__ void head_mean_elu(const float* __restrict__ agg, const float* __restrict__ bg,
                              float* __restrict__ h2) {
  unsigned t = blockIdx.x * blockDim.x + threadIdx.x;
  if (t >= (unsigned)N_NODES * GO) return;
  int n = t / GO, f = t % GO;
  float v = 0.5f * (agg[(size_t)n * (HEADS * GO) + f] +
                    agg[(size_t)n * (HEADS * GO) + GO + f]) + bg[f];
  h2[t] = v > 0.0f ? v : (expf(v) - 1.0f);   // ELU(alpha=1)
}

// ---------------------------------------------------------------------------
extern "C" void kernel_launch(void* const* d_in, const int* in_sizes, int n_in,
                              void* d_out, int out_size, void* d_ws, size_t ws_size,
                              hipStream_t stream) {
  (void)in_sizes; (void)n_in; (void)out_size; (void)ws_size;

  const float* x       = (const float*)d_in[0];
  const int*   ei      = (const int*)d_in[1];
  const int*   src     = ei;
  const int*   dst     = ei + N_EDGES;
  const float* W1      = (const float*)d_in[2];
  const float* b1      = (const float*)d_in[3];
  const float* Wg      = (const float*)d_in[4];
  const float* att_src = (const float*)d_in[5];
  const float* att_dst = (const float*)d_in[6];
  const float* bg      = (const float*)d_in[7];
  const float* W3      = (const float*)d_in[8];
  const float* b3      = (const float*)d_in[9];
  float*       out     = (float*)d_out;

  // workspace layout (floats)
  float*    h1    = (float*)d_ws;                            // N*128 (h1, later hg)
  float*    agg1  = h1   + (size_t)N_NODES * 128;            // N*128 (gcn1 agg, later gat agg)
  float*    h2    = agg1 + (size_t)N_NODES * 128;            // N*64
  float*    h3    = h2   + (size_t)N_NODES * 64;             // N*64
  float*    inv   = h3   + (size_t)N_NODES * 64;             // N     (deg -> inv-sqrt)
  float*    as_   = inv  + N_NODES;                          // N*2
  float*    ad_   = as_  + (size_t)N_NODES * HEADS;          // N*2
  unsigned* m     = (unsigned*)(ad_ + (size_t)N_NODES * HEADS); // N*2
  float*    denom = (float*)(m + (size_t)N_NODES * HEADS);   // N*2

  // zero-init accumulators (graph-capturable stream memsets)
  hipMemsetAsync(inv,   0, sizeof(float) * N_NODES, stream);
  hipMemsetAsync(agg1,  0, sizeof(float) * (size_t)N_NODES * 128, stream);
  hipMemsetAsync(m,     0, sizeof(unsigned) * (size_t)N_NODES * HEADS, stream);
  hipMemsetAsync(denom, 0, sizeof(float) * (size_t)N_NODES * HEADS, stream);
  hipMemsetAsync(out,   0, sizeof(float) * (size_t)N_NODES * OUT_DIM, stream);

  // symmetric normalization coefficients
  deg_count<<<cdiv(TOT_E, 256), 256, 0, stream>>>(dst, inv);
  inv_sqrt_k<<<cdiv(N_NODES, 256), 256, 0, stream>>>(inv);

  // ----- layer 1: GCN -----
  gemm_f32_wmma<<<N_NODES / 16, 256, 0, stream>>>(x, W1, h1, IN_DIM, HID);
  gcn_scatter<128><<<cdiv((long long)TOT_E * 32, 256), 256, 0, stream>>>(h1, inv, src, dst, agg1);
  bias_act<128, 1><<<cdiv((long long)N_NODES * 128, 256), 256, 0, stream>>>(agg1, b1);

  // ----- layer 2: GAT -----
  gemm_f32_wmma<<<N_NODES / 16, 256, 0, stream>>>(agg1, Wg, h1, HID, HEADS * GO); // hg -> h1
  attn_coeff<<<cdiv(N_NODES * HEADS, 256), 256, 0, stream>>>(h1, att_src, att_dst, as_, ad_);
  gat_edge_max<<<cdiv(TOT_E, 256), 256, 0, stream>>>(as_, ad_, src, dst, m);
  gat_edge_sum<<<cdiv(TOT_E, 256), 256, 0, stream>>>(as_, ad_, src, dst, m, denom);
  hipMemsetAsync(agg1, 0, sizeof(float) * (size_t)N_NODES * 128, stream);        // reuse as gat agg
  gat_scatter<<<cdiv((long long)TOT_E * 32, 256), 256, 0, stream>>>(h1, as_, ad_, m, denom, src, dst, agg1);
  head_mean_elu<<<cdiv((long long)N_NODES * GO, 256), 256, 0, stream>>>(agg1, bg, h2);

  // ----- layer 3: GCN -----
  gemm_f32_wmma<<<N_NODES / 16, 256, 0, stream>>>(h2, W3, h3, GO, OUT_DIM);
  gcn_scatter<64><<<cdiv((long long)TOT_E * 16, 256), 256, 0, stream>>>(h3, inv, src, dst, out);
  bias_act<64, 0><<<cdiv((long long)N_NODES * 64, 256), 256, 0, stream>>>(out, b3);
}